// GroupSortActivation_16724602651343
// MI455X (gfx1250) — compile-verified
//
#include <hip/hip_runtime.h>

// GroupSortActivation, GROUP_SIZE=2, x: (16384, 4096) f32.
// Pairwise (min,max) over adjacent element pairs -> pure streaming kernel.
// Roofline: 256MB read + 256MB write @ 23.3 TB/s => ~22us floor; VALU is
// negligible. Bulk kernel: 4 clustered global_load_b128 (TH=NT, constant
// immediate offsets), one wait, v_min_num/v_max_num pairs, 4 NT b128 stores.
// Launch geometry guarantees the bulk kernel is always in-bounds; a pair-
// granularity tail kernel covers any remainder (none for this shape).

typedef float v4f __attribute__((ext_vector_type(4)));
typedef float v2f __attribute__((ext_vector_type(2)));

#define BLOCK  256
#define CHUNKS 4

// Each block handles BLOCK*CHUNKS consecutive float4s; each thread handles
// CHUNKS float4s at constant offsets k*BLOCK from its base -> the k*4096-byte
// deltas become instruction immediate offsets, and loads cluster.
__global__ void __launch_bounds__(BLOCK)
group_sort2_bulk(const v4f* __restrict__ in, v4f* __restrict__ out) {
    const int base = blockIdx.x * (BLOCK * CHUNKS) + threadIdx.x;

    v4f v[CHUNKS];
#pragma unroll
    for (int k = 0; k < CHUNKS; ++k) {
        v[k] = __builtin_nontemporal_load(&in[base + k * BLOCK]);
    }

#pragma unroll
    for (int k = 0; k < CHUNKS; ++k) {
        v4f r;
        r.x = fminf(v[k].x, v[k].y);
        r.y = fmaxf(v[k].x, v[k].y);
        r.z = fminf(v[k].z, v[k].w);
        r.w = fmaxf(v[k].z, v[k].w);
        __builtin_nontemporal_store(r, &out[base + k * BLOCK]);
    }
}

// Tail: one thread per remaining pair (float2 granularity; n is always even).
__global__ void __launch_bounds__(BLOCK)
group_sort2_tail(const v2f* __restrict__ in, v2f* __restrict__ out,
                 int startPair, int nPair) {
    const int i = startPair + blockIdx.x * BLOCK + threadIdx.x;
    if (i < nPair) {
        v2f p = in[i];
        v2f r;
        r.x = fminf(p.x, p.y);
        r.y = fmaxf(p.x, p.y);
        out[i] = r;
    }
}

extern "C" void kernel_launch(void* const* d_in, const int* in_sizes, int n_in,
                              void* d_out, int out_size, void* d_ws, size_t ws_size,
                              hipStream_t stream) {
    (void)n_in; (void)d_ws; (void)ws_size; (void)out_size;

    const int n     = in_sizes[0];        // 67,108,864 floats (even)
    const int nPair = n / 2;              // 33,554,432 pairs
    const int n4    = n / 4;              // 16,777,216 float4

    const int perBlock   = BLOCK * CHUNKS;        // 1024 float4 per block
    const int bulkBlocks = n4 / perBlock;         // 16384 for this shape
    const int full4      = bulkBlocks * perBlock; // float4s covered by bulk

    if (bulkBlocks > 0) {
        group_sort2_bulk<<<bulkBlocks, BLOCK, 0, stream>>>(
            (const v4f*)d_in[0], (v4f*)d_out);
    }

    const int startPair = full4 * 2;
    const int remPairs  = nPair - startPair;      // 0 for this shape
    if (remPairs > 0) {
        const int tailBlocks = (remPairs + BLOCK - 1) / BLOCK;
        group_sort2_tail<<<tailBlocks, BLOCK, 0, stream>>>(
            (const v2f*)d_in[0], (v2f*)d_out, startPair, nPair);
    }
}